// HeteroGraphClassifier_88235808129790
// MI455X (gfx1250) — compile-verified
//
#include <hip/hip_runtime.h>
#include <hip/hip_bf16.h>

// ---------------- problem constants (match reference) ----------------
#define LNUM   8
#define BATCH  128
#define NL     32
#define NPN    256            // protein nodes per graph
#define N_LIG  (BATCH * NL)   // 4096
#define N_PROT (BATCH * NPN)  // 32768
#define EPLG   1024           // protein->ligand edges per graph
#define ELLG   256            // ligand->ligand edges per graph
#define FIN    128
#define HD     256
#define OUTD   2

typedef __attribute__((ext_vector_type(2))) float v2f;
typedef __attribute__((ext_vector_type(8))) float v8f;

// ---------------- kernel 0: zero scratch region ----------------
__global__ void k_zero(float* __restrict__ p, int n) {
    int i = blockIdx.x * 256 + threadIdx.x;
    if (i < n) p[i] = 0.0f;
}

// ---------------- kernel 1: degree counting (segment_sum of ones) ----------------
// idx: [L, E] global node ids (already graph-offset). deg: [L, n_per_layer].
__global__ void k_deg(const int* __restrict__ idx, float* __restrict__ deg,
                      int e_per_layer, int n_per_layer, int total) {
    int t = blockIdx.x * 256 + threadIdx.x;
    if (t >= total) return;
    int l = t / e_per_layer;
    atomicAdd(&deg[l * n_per_layer + idx[t]], 1.0f);
}

// ---------------- kernel 2: per-(layer,graph) weighted edge aggregation ----------
// v[(l*B+g)*FIN + f] = (1/NL) * sum_e w_e * feat[l, src_e, f]
// w_e = rsqrt(max(deg_src[src],1)) * rsqrt(max(deg_dst[dst],1))
// block = 128 threads (thread = feature), grid = L*B blocks.
__global__ __launch_bounds__(FIN)
void k_edge_accum(const float* __restrict__ feat,   // [L, n_src, FIN]
                  const int* __restrict__ src,      // [L, B*epg]
                  const int* __restrict__ dst,      // [L, B*epg]
                  const float* __restrict__ deg_s,  // [L, n_src]
                  const float* __restrict__ deg_d,  // [L, N_LIG]
                  float* __restrict__ vout,         // [L, B, FIN]
                  int n_src, int epg) {
    int l = blockIdx.x / BATCH;
    int g = blockIdx.x % BATCH;
    int f = threadIdx.x;
    const int   ebase = l * (BATCH * epg) + g * epg;
    const float* fb   = feat  + (size_t)l * n_src * FIN;
    const float* dsb  = deg_s + (size_t)l * n_src;
    const float* ddb  = deg_d + (size_t)l * N_LIG;
    float acc = 0.0f;
    for (int e = 0; e < epg; ++e) {
        int s = src[ebase + e];
        int d = dst[ebase + e];
        float w = rsqrtf(fmaxf(dsb[s], 1.0f)) * rsqrtf(fmaxf(ddb[d], 1.0f));
        acc += w * fb[(size_t)s * FIN + f];
    }
    vout[((size_t)l * BATCH + g) * FIN + f] = acc * (1.0f / (float)NL);
}

// ---------------- kernel 3: WMMA projection  h = v_pl@W_pl + v_ll@W_ll + b ------
// One wave per 16x16 tile of h[l] (M=graphs=128 -> 8 tiles, N=H=256 -> 16 tiles).
// V_WMMA_F32_16X16X4_F32:
//   A 16x4 f32:  lane L, reg r: M = L%16, K = (L/16)*2 + r
//   B 4x16 f32:  lane L, reg r: N = L%16, K = (L/16)*2 + r
//   C/D 16x16:   lane L, reg r: N = L%16, M = r + 8*(L/16)
__global__ __launch_bounds__(128)
void k_gemm_wmma(const float* __restrict__ vpl,  // [L,B,FIN]
                 const float* __restrict__ vll,  // [L,B,FIN]
                 const float* __restrict__ Wpl,  // [L,FIN,HD]
                 const float* __restrict__ Wll,  // [L,FIN,HD]
                 const float* __restrict__ bpl,  // [L,HD]
                 const float* __restrict__ bll,  // [L,HD]
                 float* __restrict__ hbuf) {     // [L,B,HD]
    const int MT = BATCH / 16;   // 8
    const int NT = HD / 16;      // 16
    int wave = threadIdx.x >> 5;
    int lane = threadIdx.x & 31;
    int tile = blockIdx.x * 4 + wave;            // grid = L*MT*NT/4 = 256
    int l   = tile / (MT * NT);
    int rem = tile % (MT * NT);
    int m0  = (rem / NT) * 16;
    int n0  = (rem % NT) * 16;
    int ln  = lane & 15;
    int hi  = lane >> 4;                         // 0 or 1
    int n   = n0 + ln;

    float bias = bpl[l * HD + n] + bll[l * HD + n];
    v8f c;
#pragma unroll
    for (int i = 0; i < 8; ++i) c[i] = bias;

    const float* Apl = vpl + ((size_t)l * BATCH + (m0 + ln)) * FIN;
    const float* All = vll + ((size_t)l * BATCH + (m0 + ln)) * FIN;
    const float* Bpl = Wpl + (size_t)l * FIN * HD + n;
    const float* Bll = Wll + (size_t)l * FIN * HD + n;

    for (int kk = 0; kk < FIN; kk += 4) {
        int kb = kk + hi * 2;
        v2f a, b;
        a[0] = Apl[kb];          a[1] = Apl[kb + 1];
        b[0] = Bpl[(size_t)kb * HD]; b[1] = Bpl[(size_t)(kb + 1) * HD];
        c = __builtin_amdgcn_wmma_f32_16x16x4_f32(false, a, false, b,
                                                  (short)0, c, false, false);
    }
    for (int kk = 0; kk < FIN; kk += 4) {
        int kb = kk + hi * 2;
        v2f a, b;
        a[0] = All[kb];          a[1] = All[kb + 1];
        b[0] = Bll[(size_t)kb * HD]; b[1] = Bll[(size_t)(kb + 1) * HD];
        c = __builtin_amdgcn_wmma_f32_16x16x4_f32(false, a, false, b,
                                                  (short)0, c, false, false);
    }

#pragma unroll
    for (int r = 0; r < 8; ++r) {
        int M = m0 + r + hi * 8;
        hbuf[((size_t)l * BATCH + M) * HD + n] = c[r];
    }
}

// ---------------- kernel 4: fused FC (concat over L) + sigmoid ------------------
// out[b,o] = sigmoid( sum_{l,h} hbuf[l,b,h] * W_fc[(l*HD+h)*2 + o] + b_fc[o] )
__global__ __launch_bounds__(256)
void k_fc_sigmoid(const float* __restrict__ hbuf, const float* __restrict__ Wfc,
                  const float* __restrict__ bfc, float* __restrict__ out) {
    int t = blockIdx.x * 256 + threadIdx.x;       // 0..255
    int b = t >> 1, o = t & 1;
    float acc = bfc[o];
    for (int l = 0; l < LNUM; ++l) {
        const float* hrow = hbuf + ((size_t)l * BATCH + b) * HD;
        const float* w    = Wfc + (size_t)(l * HD) * OUTD + o;
        for (int h = 0; h < HD; ++h) acc += hrow[h] * w[(size_t)h * OUTD];
    }
    out[b * OUTD + o] = 1.0f / (1.0f + __expf(-acc));
}

// ---------------- host-side launcher ----------------
extern "C" void kernel_launch(void* const* d_in, const int* in_sizes, int n_in,
                              void* d_out, int out_size, void* d_ws, size_t ws_size,
                              hipStream_t stream) {
    const float* feat_lig  = (const float*)d_in[0];   // [L, N_LIG, FIN]
    const float* feat_prot = (const float*)d_in[1];   // [L, N_PROT, FIN]
    const int*   src_pl    = (const int*)d_in[2];     // [L, E_PL]
    const int*   dst_pl    = (const int*)d_in[3];
    const int*   src_ll    = (const int*)d_in[4];
    const int*   dst_ll    = (const int*)d_in[5];
    const float* W_pl      = (const float*)d_in[6];   // [L, FIN, HD]
    const float* b_pl      = (const float*)d_in[7];   // [L, HD]
    const float* W_ll      = (const float*)d_in[8];
    const float* b_ll      = (const float*)d_in[9];
    const float* W_fc      = (const float*)d_in[10];  // [L*HD, OUTD]
    const float* b_fc      = (const float*)d_in[11];  // [OUTD]
    float*       out       = (float*)d_out;          // [BATCH, OUTD]

    // workspace layout (floats)
    float* ws = (float*)d_ws;
    float* deg_sp = ws;                                   // L*N_PROT = 262144
    float* deg_dp = deg_sp + LNUM * N_PROT;               // L*N_LIG  = 32768
    float* deg_sl = deg_dp + LNUM * N_LIG;                // 32768
    float* deg_dl = deg_sl + LNUM * N_LIG;                // 32768
    float* v_pl   = deg_dl + LNUM * N_LIG;                // L*B*FIN  = 131072
    float* v_ll   = v_pl   + LNUM * BATCH * FIN;          // 131072
    float* hbuf   = v_ll   + LNUM * BATCH * FIN;          // L*B*HD   = 262144

    const int deg_total = LNUM * (N_PROT + 3 * N_LIG);    // 360448
    k_zero<<<(deg_total + 255) / 256, 256, 0, stream>>>(deg_sp, deg_total);

    const int E_PL = BATCH * EPLG;                        // 131072
    const int E_LL = BATCH * ELLG;                        // 32768
    k_deg<<<(LNUM * E_PL + 255) / 256, 256, 0, stream>>>(src_pl, deg_sp, E_PL, N_PROT, LNUM * E_PL);
    k_deg<<<(LNUM * E_PL + 255) / 256, 256, 0, stream>>>(dst_pl, deg_dp, E_PL, N_LIG,  LNUM * E_PL);
    k_deg<<<(LNUM * E_LL + 255) / 256, 256, 0, stream>>>(src_ll, deg_sl, E_LL, N_LIG,  LNUM * E_LL);
    k_deg<<<(LNUM * E_LL + 255) / 256, 256, 0, stream>>>(dst_ll, deg_dl, E_LL, N_LIG,  LNUM * E_LL);

    // weighted edge aggregation folded with per-graph mean (linear GraphConv)
    k_edge_accum<<<LNUM * BATCH, FIN, 0, stream>>>(feat_prot, src_pl, dst_pl,
                                                   deg_sp, deg_dp, v_pl, N_PROT, EPLG);
    k_edge_accum<<<LNUM * BATCH, FIN, 0, stream>>>(feat_lig, src_ll, dst_ll,
                                                   deg_sl, deg_dl, v_ll, N_LIG, ELLG);

    // WMMA fp32 projection: h[l] = v_pl[l]@W_pl[l] + v_ll[l]@W_ll[l] + b
    k_gemm_wmma<<<(LNUM * (BATCH / 16) * (HD / 16)) / 4, 128, 0, stream>>>(
        v_pl, v_ll, W_pl, W_ll, b_pl, b_ll, hbuf);

    // fused concat + FC + sigmoid
    k_fc_sigmoid<<<1, 256, 0, stream>>>(hbuf, W_fc, b_fc, out);
}